// LossYOLOv1_41661182771226
// MI455X (gfx1250) — compile-verified
//
#include <hip/hip_runtime.h>
#include <math.h>

#ifndef __has_builtin
#define __has_builtin(x) 0
#endif

#define GRIDSZ 7
#define NCLS 20
#define DCH 30
#define BATCH 8192
#define NCELL (BATCH * GRIDSZ * GRIDSZ) /* 401408 */
#define TPB 256
#define NBLK (NCELL / TPB)              /* 1568 exactly */
#define CHUNK_F4 (TPB * DCH / 4)        /* 1920 float4 per array per block */

typedef float v2f __attribute__((ext_vector_type(2)));
typedef float v8f __attribute__((ext_vector_type(8)));
typedef int v4i_vs __attribute__((vector_size(16)));
typedef __attribute__((address_space(1))) v4i_vs* gptr_v4i;
typedef __attribute__((address_space(3))) v4i_vs* lptr_v4i;

__device__ __forceinline__ float bce_f(float x, float t) {
  // max(x,0) - x*t + log1p(exp(-|x|))
  return fmaxf(x, 0.0f) - x * t + log1pf(expf(-fabsf(x)));
}

__device__ __forceinline__ float sigm_f(float x) {
  return 1.0f / (1.0f + expf(-x));
}

__device__ __forceinline__ float iou_f(float al, float at, float ar, float ab,
                                       float bl, float bt, float br, float bb) {
  float lx = fmaxf(al, bl), ly = fmaxf(at, bt);
  float rx = fminf(ar, br), ry = fminf(ab, bb);
  float w = fmaxf(rx - lx, 0.0f), h = fmaxf(ry - ly, 0.0f);
  float inter = w * h;
  float aa = (ar - al) * (ab - at);
  float ba = (br - bl) * (bb - bt);
  return inter / (aa + ba - inter + 1e-7f);
}

__global__ __launch_bounds__(TPB) void yolo_loss_cells(
    const float* __restrict__ p, const float* __restrict__ g,
    float* __restrict__ block_sums) {
  __shared__ __align__(16) float lp[TPB * DCH];
  __shared__ __align__(16) float lg[TPB * DCH];
  __shared__ float wacc[TPB / 32];

  const int tid = threadIdx.x;
  const size_t base = (size_t)blockIdx.x * (TPB * DCH);
  const float4* gp = reinterpret_cast<const float4*>(p + base);
  const float4* gg = reinterpret_cast<const float4*>(g + base);

  // ---- Stage 30720B of p and of g into LDS, fully coalesced B128 ----
#if __has_builtin(__builtin_amdgcn_global_load_async_to_lds_b128)
  for (int i = tid; i < CHUNK_F4; i += TPB) {
    __builtin_amdgcn_global_load_async_to_lds_b128(
        (gptr_v4i)(gp + i), (lptr_v4i)(&lp[4 * i]), 0, 0);
    __builtin_amdgcn_global_load_async_to_lds_b128(
        (gptr_v4i)(gg + i), (lptr_v4i)(&lg[4 * i]), 0, 0);
  }
#if __has_builtin(__builtin_amdgcn_s_wait_asynccnt)
  __builtin_amdgcn_s_wait_asynccnt(0);
#else
  asm volatile("s_wait_asynccnt 0" ::: "memory");
#endif
#else
  for (int i = tid; i < CHUNK_F4; i += TPB) {
    float4 vp = gp[i];
    float4 vg = gg[i];
    *reinterpret_cast<float4*>(&lp[4 * i]) = vp;
    *reinterpret_cast<float4*>(&lg[4 * i]) = vg;
  }
#endif
  __syncthreads();

  // ---- Per-cell loss (stride-30 LDS reads: conflict-free across 32 lanes) ----
  const float* cp = &lp[tid * DCH];
  const float* cg = &lg[tid * DCH];
  const int cell = blockIdx.x * TPB + tid;
  const int ci = cell % (GRIDSZ * GRIDSZ);
  const float row = (float)(ci / GRIDSZ);
  const float col = (float)(ci % GRIDSZ);

  // class loss over 20 channels
  float cls_cell = 0.0f;
#pragma unroll
  for (int k = 0; k < NCLS; ++k) cls_cell += bce_f(cp[10 + k], cg[10 + k]);

  const float pc0 = cp[8], pc1 = cp[9];
  const float neg_cell = bce_f(pc0, 0.0f) + bce_f(pc1, 0.0f);

  float ox[2], oy[2], ow[2], oh[2];
  float gx[2], gy[2], gw[2], gh[2];
#pragma unroll
  for (int b = 0; b < 2; ++b) {
    ox[b] = sigm_f(cp[4 * b + 0]);
    oy[b] = sigm_f(cp[4 * b + 1]);
    ow[b] = cp[4 * b + 2];
    oh[b] = cp[4 * b + 3];
    gx[b] = cg[4 * b + 0];
    gy[b] = cg[4 * b + 1];
    gw[b] = cg[4 * b + 2];
    gh[b] = cg[4 * b + 3];
  }

  float pl[2], pt[2], pr[2], pb[2], gl[2], gt[2], gr[2], gb[2];
#pragma unroll
  for (int b = 0; b < 2; ++b) {
    float cx = (ox[b] + col) / (float)GRIDSZ;
    float cy = (oy[b] + row) / (float)GRIDSZ;
    pl[b] = cx - ow[b] * 0.5f;
    pt[b] = cy - oh[b] * 0.5f;
    pr[b] = cx + ow[b] * 0.5f;
    pb[b] = cy + oh[b] * 0.5f;
    float gcx = (gx[b] + col) / (float)GRIDSZ;
    float gcy = (gy[b] + row) / (float)GRIDSZ;
    gl[b] = gcx - gw[b] * 0.5f;
    gt[b] = gcy - gh[b] * 0.5f;
    gr[b] = gcx + gw[b] * 0.5f;
    gb[b] = gcy + gh[b] * 0.5f;
  }

  int ind[2];
#pragma unroll
  for (int j = 0; j < 2; ++j) {
    float i0 = iou_f(pl[0], pt[0], pr[0], pb[0], gl[j], gt[j], gr[j], gb[j]);
    float i1 = iou_f(pl[1], pt[1], pr[1], pb[1], gl[j], gt[j], gr[j], gb[j]);
    ind[j] = (i1 > i0) ? 1 : 0;  // jnp.argmax tie -> first index
  }

  auto box_l = [&](int i, int j) {
    float dx = ox[i] - gx[j];
    float dy = oy[i] - gy[j];
    float dw = sqrtf(fabsf(ow[i])) - sqrtf(gw[j]);
    float dh = sqrtf(fabsf(oh[i])) - sqrtf(gh[j]);
    return dx * dx + dy * dy + dw * dw + dh * dh;
  };

  const bool same_g = (gx[0] == gx[1]) && (gy[0] == gy[1]) &&
                      (gw[0] == gw[1]) && (gh[0] == gh[1]);
  const bool same_ind = (ind[0] == ind[1]);

  const float lossA = box_l(ind[0], 0);
  const float box_cell =
      same_g ? lossA
             : (same_ind ? (box_l(0, 0) + box_l(1, 1))
                         : (lossA + box_l(ind[1], 1)));

  const float confA = bce_f(ind[1] ? pc1 : pc0, 1.0f);
  const float conf_cell =
      same_g ? confA : (bce_f(pc0, 1.0f) + bce_f(pc1, 1.0f));

  const float conf_g = cg[8];
  float L = 0.0f;
  if (conf_g > 0.0f) L += 5.0f * box_cell + conf_cell + cls_cell;
  if (conf_g == 0.0f) L += 0.5f * neg_cell;

  // ---- Wave reduction: exact f32 via V_WMMA_F32_16X16X4_F32 with A=ones ----
  float lanesum;
#if __has_builtin(__builtin_amdgcn_wmma_f32_16x16x4_f32)
  {
    v2f a;
    a[0] = 1.0f;
    a[1] = 1.0f;                 // A(16x4) = all ones
    v2f b;
    b[0] = L;                    // B(4x16): K0 <- lanes 0-15, K2 <- lanes 16-31
    b[1] = 0.0f;                 // K1/K3 = 0
    v8f c = {};
    c = __builtin_amdgcn_wmma_f32_16x16x4_f32(false, a, false, b, (short)0, c,
                                              false, false);
    lanesum = c[0];              // column sum L[n] + L[n+16], duplicated per row
    lanesum += __shfl_xor(lanesum, 8, 32);
    lanesum += __shfl_xor(lanesum, 4, 32);
    lanesum += __shfl_xor(lanesum, 2, 32);
    lanesum += __shfl_xor(lanesum, 1, 32);
  }
#else
  lanesum = L;
  for (int off = 16; off > 0; off >>= 1)
    lanesum += __shfl_xor(lanesum, off, 32);
#endif

  if ((tid & 31) == 0) wacc[tid >> 5] = lanesum;
  __syncthreads();
  if (tid == 0) {
    float s = 0.0f;
#pragma unroll
    for (int w = 0; w < TPB / 32; ++w) s += wacc[w];
    block_sums[blockIdx.x] = s;
  }
}

__global__ __launch_bounds__(TPB) void yolo_loss_finalize(
    const float* __restrict__ block_sums, float* __restrict__ out, int n) {
  __shared__ float s[TPB];
  float v = 0.0f;
  for (int i = threadIdx.x; i < n; i += TPB) v += block_sums[i];
  s[threadIdx.x] = v;
  __syncthreads();
  for (int off = TPB / 2; off > 0; off >>= 1) {
    if (threadIdx.x < off) s[threadIdx.x] += s[threadIdx.x + off];
    __syncthreads();
  }
  if (threadIdx.x == 0) out[0] = s[0] * (1.0f / (float)BATCH);
}

extern "C" void kernel_launch(void* const* d_in, const int* in_sizes, int n_in,
                              void* d_out, int out_size, void* d_ws,
                              size_t ws_size, hipStream_t stream) {
  (void)in_sizes; (void)n_in; (void)out_size; (void)ws_size;
  const float* p = (const float*)d_in[0];
  const float* g = (const float*)d_in[1];
  float* out = (float*)d_out;
  float* bsums = (float*)d_ws;  // NBLK * 4 bytes of scratch

  yolo_loss_cells<<<NBLK, TPB, 0, stream>>>(p, g, bsums);
  yolo_loss_finalize<<<1, TPB, 0, stream>>>(bsums, out, NBLK);
}